// MaskedQueryAndGroup_26225070309536
// MI455X (gfx1250) — compile-verified
//
#include <hip/hip_runtime.h>
#include <hip/hip_bf16.h>

// Problem constants (match reference setup_inputs)
#define BATCH   4
#define NPT     2048      // npoint (queries)
#define NSUP    8192      // N (support points)
#define CFEAT   64        // feature channels
#define NS      32        // nsample
#define RAD2    0.01f     // radius^2
#define INV_RAD 10.0f     // 1/radius

typedef __attribute__((ext_vector_type(2))) float v2f;
typedef __attribute__((ext_vector_type(8))) float v8f;

// ---------------------------------------------------------------------------
// Kernel 1: masked *ordered* ball query via V_WMMA_F32_16X16X4_F32.
// One wave = 16 queries. Each iteration: one WMMA produces a 16x16 tile of
// q.s dot products; d2 = |q|^2 + |s|^2 - 2*dot; 8 wave32 ballots convert the
// in-ball predicate into per-query 16-bit masks in ascending support-index
// order; lanes 0..15 compact the first 32 indices per query into LDS.
// ---------------------------------------------------------------------------
__global__ __launch_bounds__(128) void ballquery_wmma_kernel(
    const float* __restrict__ Q,          // (B,3,NPT)
    const float* __restrict__ S,          // (B,3,NSUP)
    const unsigned char* __restrict__ QM, // (B,NPT)  bool
    const unsigned char* __restrict__ SM, // (B,NSUP) bool
    int*   __restrict__ idx_out,          // (B,NPT,NS) in workspace
    float* __restrict__ mask_out)         // (B,NPT,NS) tail of d_out
{
    __shared__ int lidx[4 * 16 * NS];     // 4 waves * 16 queries * 32 idx = 8KB

    const int lane = threadIdx.x & 31;
    const int wave = threadIdx.x >> 5;
    const int col  = lane & 15;           // N within 16-wide support chunk
    const int hi   = lane >> 4;           // which half of the wave

    const int gq0 = blockIdx.x * 64 + wave * 16;  // first query of this wave
    const int b   = gq0 / NPT;
    const int p0  = gq0 % NPT;            // 64 | NPT, so whole tile same batch

    const float* Qb  = Q  + (size_t)b * 3 * NPT;
    const float* Sb  = S  + (size_t)b * 3 * NSUP;
    const unsigned char* SMb = SM + (size_t)b * NSUP;

    // ---- A matrix (16x4 f32): lane<16 holds (K0,K1)=(qx,qy) of row M=col,
    //      lane>=16 holds (K2,K3)=(qz,0) of row M=col. -------------------
    const int pm = p0 + col;
    const float qx = Qb[pm];
    const float qy = Qb[NPT + pm];
    const float qz = Qb[2 * NPT + pm];
    v2f a;
    a.x = hi ? qz : qx;
    a.y = hi ? 0.0f : qy;

    // |q|^2 for the 8 rows this lane's accumulator covers (M = v + 8*hi)
    float qn[8];
#pragma unroll
    for (int v = 0; v < 8; ++v) {
        const int mq = p0 + hi * 8 + v;
        const float x = Qb[mq], y = Qb[NPT + mq], z = Qb[2 * NPT + mq];
        qn[v] = x * x + y * y + z * z;
    }

    // Selection-lane state (lanes 0..15 each own query m = lane)
    int  cnt = 0;
    bool qm  = false;
    int* myl = nullptr;
    if (lane < 16) {
        qm  = QM[(size_t)b * NPT + p0 + lane] != 0;
        myl = &lidx[(wave * 16 + lane) * NS];
    }

    for (int chunk = 0; chunk < NSUP / 16; ++chunk) {
        const int n = chunk * 16 + col;
        const float sx = Sb[n];
        const float sy = Sb[NSUP + n];
        const float sz = Sb[2 * NSUP + n];

        // ---- B matrix (4x16 f32): lane<16 holds (K0,K1)=(sx,sy) of col N,
        //      lane>=16 holds (K2,K3)=(sz,0) of col N. ---------------------
        v2f bm;
        bm.x = hi ? sz : sx;
        bm.y = hi ? 0.0f : sy;

        const float sn = sx * sx + sy * sy + sz * sz;      // |s|^2 for N=col
        const bool  sm = SMb[n] != 0;

        v8f acc = {0.f, 0.f, 0.f, 0.f, 0.f, 0.f, 0.f, 0.f};
        // D[M][N] = q . s   (K=4, pad element is 0)
        acc = __builtin_amdgcn_wmma_f32_16x16x4_f32(
            false, a, false, bm, (short)0, acc, false, false);

        // 8 ballots: bl[v] bits[15:0] = within(M=v,  N=0..15)
        //            bl[v] bits[31:16]= within(M=v+8,N=0..15)
        unsigned bl[8];
#pragma unroll
        for (int v = 0; v < 8; ++v) {
            const float d2 = qn[v] + sn - 2.0f * acc[v];
            bl[v] = (unsigned)__builtin_amdgcn_ballot_w32(sm && (d2 <= RAD2));
        }

        // Lanes 0..15: append this chunk's in-ball indices (ascending order)
        if (lane < 16 && qm && cnt < NS) {
            const unsigned m8 = lane & 7u;
            const unsigned lo4 = (m8 & 2u) ? ((m8 & 1u) ? bl[3] : bl[2])
                                           : ((m8 & 1u) ? bl[1] : bl[0]);
            const unsigned hi4 = (m8 & 2u) ? ((m8 & 1u) ? bl[7] : bl[6])
                                           : ((m8 & 1u) ? bl[5] : bl[4]);
            const unsigned r  = (m8 & 4u) ? hi4 : lo4;
            unsigned mm = (lane < 8) ? (r & 0xFFFFu) : (r >> 16);
            while (mm && cnt < NS) {
                const int bit = __builtin_ctz(mm);
                myl[cnt++] = chunk * 16 + bit;
                mm &= mm - 1u;
            }
        }

        // Early exit once every (unmasked) query has its 32 neighbors
        const bool notdone = (lane < 16) && qm && (cnt < NS);
        if (__builtin_amdgcn_ballot_w32(notdone) == 0) break;
    }

    // Epilogue: pad with first neighbor (0 if none), emit idx + mask
    if (lane < 16) {
        const int gq = gq0 + lane;
        const int first = (cnt > 0) ? myl[0] : 0;
        const size_t base = (size_t)gq * NS;
        for (int j = 0; j < NS; ++j) {
            idx_out[base + j]  = (j < cnt) ? myl[j] : first;
            mask_out[base + j] = (j < cnt) ? 1.0f : 0.0f;
        }
    }
}

// ---------------------------------------------------------------------------
// Kernel 2: gather + write new_features (B, 3+C, NPT, NS).
// Thread per (b,q,j): 67 coalesced channel writes (consecutive j per lane),
// feature gathers hit L2 (features = 8MB << 192MB L2).
// ---------------------------------------------------------------------------
__global__ __launch_bounds__(256) void group_gather_kernel(
    const float* __restrict__ Q,   // (B,3,NPT)
    const float* __restrict__ S,   // (B,3,NSUP)
    const float* __restrict__ F,   // (B,C,NSUP)
    const int*   __restrict__ idx, // (B,NPT,NS)
    float* __restrict__ out)       // (B,67,NPT,NS)
{
    const int t = blockIdx.x * 256 + threadIdx.x;   // < B*NPT*NS = 262144
    const int j = t & (NS - 1);
    const int q = (t >> 5) & (NPT - 1);
    const int b = t >> 16;                           // NPT*NS = 1<<16
    const int i = idx[t];

    const size_t chs   = (size_t)NPT * NS;                       // 65536
    const size_t obase = (size_t)b * 67 * chs + (size_t)q * NS + j;

    const float* Qb = Q + (size_t)b * 3 * NPT;
    const float* Sb = S + (size_t)b * 3 * NSUP;

#pragma unroll
    for (int d = 0; d < 3; ++d) {
        out[obase + (size_t)d * chs] =
            (Sb[(size_t)d * NSUP + i] - Qb[(size_t)d * NPT + q]) * INV_RAD;
    }

    const float* Fb = F + (size_t)b * CFEAT * NSUP;
#pragma unroll 8
    for (int c = 0; c < CFEAT; ++c) {
        out[obase + (size_t)(3 + c) * chs] = Fb[(size_t)c * NSUP + i];
    }
}

extern "C" void kernel_launch(void* const* d_in, const int* in_sizes, int n_in,
                              void* d_out, int out_size, void* d_ws, size_t ws_size,
                              hipStream_t stream) {
    (void)in_sizes; (void)n_in; (void)out_size; (void)ws_size;

    const float*         Q  = (const float*)d_in[0];          // (4,3,2048)
    const float*         S  = (const float*)d_in[1];          // (4,3,8192)
    const unsigned char* QM = (const unsigned char*)d_in[2];  // (4,2048) bool
    const unsigned char* SM = (const unsigned char*)d_in[3];  // (4,8192) bool
    const float*         F  = (const float*)d_in[4];          // (4,64,8192)

    float* out      = (float*)d_out;
    int*   ws_idx   = (int*)d_ws;                             // 262144 ints = 1MB
    float* mask_out = out + (size_t)BATCH * 67 * NPT * NS;    // idx_mask tail

    // 1 wave = 16 queries, 4 waves/block => 64 queries/block => 128 blocks
    ballquery_wmma_kernel<<<(BATCH * NPT) / 64, 128, 0, stream>>>(
        Q, S, QM, SM, ws_idx, mask_out);

    // one thread per (b, q, j)
    group_gather_kernel<<<(BATCH * NPT * NS) / 256, 256, 0, stream>>>(
        Q, S, F, ws_idx, out);
}